// DiagonalQuadratic_40200893891173
// MI455X (gfx1250) — compile-verified
//
#include <hip/hip_runtime.h>

typedef float v2f __attribute__((ext_vector_type(2)));
typedef float v8f __attribute__((ext_vector_type(8)));

// y[i] = sum_k Q[k]*x[i,k]^2 + sum_k b[k]*x[i,k] + c
//
// One wave per 16-row tile (max occupancy for latency hiding). Lane l
// accumulates row (l&15); lanes 0-15 take columns [0,K/2), lanes 16-31 take
// [K/2,K). Q and b (4KB) are staged in LDS once per block so the hot loop
// issues exactly one global b128 (x payload) per 4 elements; Q/b come from
// the independent DS pipe (ds_load_b128, 2-address broadcast, conflict-free).
//
// The lane<->lane+16 reduction is one V_WMMA_F32_16X16X4_F32 with B == ones:
// f32 A-layout puts lane m V0/V1 and lane m+16 V0/V1 into row m's K slots,
// so D[m][n] = s(lane m) + s(lane m+16) for every n (B layout irrelevant
// since every B element is 1.0).
__global__ __launch_bounds__(256) void DiagonalQuadratic_kernel(
    const float* __restrict__ x,
    const float* __restrict__ Q,
    const float* __restrict__ b,
    const float* __restrict__ c,
    float* __restrict__ out,
    int nTiles, int K)
{
    extern __shared__ float smem[];          // [0,K): Q   [K,2K): b
    float* __restrict__ lq = smem;
    float* __restrict__ lb = smem + K;

    const int tid  = threadIdx.x;
    const int lane = tid & 31;
    const int wave = tid >> 5;
    const int tile = blockIdx.x * 8 + wave;

    // Stage Q/b into LDS (one-time, coalesced).
    for (int i = tid; i < K; i += 256) {
        lq[i] = Q[i];
        lb[i] = b[i];
    }
    __syncthreads();

    if (tile >= nTiles) return;              // uniform across the wave

    const int halfK   = K >> 1;              // 256 for K=512
    const int row     = tile * 16 + (lane & 15);
    const int colBase = (lane >> 4) * halfK; // 0 or K/2

    const float* __restrict__ xr = x + (size_t)row * (size_t)K + colBase;
    const float* __restrict__ qr = lq + colBase;
    const float* __restrict__ br = lb + colBase;

    // 4 independent accumulator chains (compiler pairs them into v_pk_fma_f32).
    float s0 = 0.0f, s1 = 0.0f, s2 = 0.0f, s3 = 0.0f;
#pragma unroll 4
    for (int k = 0; k < halfK; k += 4) {
        float4 xv = *(const float4*)(xr + k);   // global_load_b128 (stream)
        float4 qv = *(const float4*)(qr + k);   // ds_load_b128 (broadcast)
        float4 bv = *(const float4*)(br + k);   // ds_load_b128 (broadcast)
        float t0 = __builtin_fmaf(qv.x, xv.x, bv.x);
        float t1 = __builtin_fmaf(qv.y, xv.y, bv.y);
        float t2 = __builtin_fmaf(qv.z, xv.z, bv.z);
        float t3 = __builtin_fmaf(qv.w, xv.w, bv.w);
        s0 = __builtin_fmaf(xv.x, t0, s0);
        s1 = __builtin_fmaf(xv.y, t1, s1);
        s2 = __builtin_fmaf(xv.z, t2, s2);
        s3 = __builtin_fmaf(xv.w, t3, s3);
    }
    const float s = (s0 + s1) + (s2 + s3);

    // Cross-half reduction via WMMA (D[m][n] = s(lane m) + s(lane m+16)).
    v2f a;    a[0] = s;    a[1] = 0.0f;
    v2f ones; ones[0] = 1.0f; ones[1] = 1.0f;
    v8f cz = {};
    v8f d = __builtin_amdgcn_wmma_f32_16x16x4_f32(
        /*neg_a=*/false, a, /*neg_b=*/false, ones,
        /*c_mod=*/(short)0, cz, /*reuse_a=*/false, /*reuse_b=*/false);

    // C/D layout: VGPR r, lanes 0-15 -> M=r; lanes 16-31 -> M=r+8.
    // All columns identical, so d[r] at any lane<16 == rowsum(r),
    // and at any lane>=16 == rowsum(r+8).
    const int r = lane & 7;
    float y = d[0];
    y = (r == 1) ? d[1] : y;
    y = (r == 2) ? d[2] : y;
    y = (r == 3) ? d[3] : y;
    y = (r == 4) ? d[4] : y;
    y = (r == 5) ? d[5] : y;
    y = (r == 6) ? d[6] : y;
    y = (r == 7) ? d[7] : y;

    const float cc = c[0];
    const bool act = (lane < 8) || (lane >= 16 && lane < 24);
    const int  m   = (lane < 8) ? lane : (lane - 8);   // rows 0-7 / 8-15
    if (act) out[tile * 16 + m] = y + cc;
}

extern "C" void kernel_launch(void* const* d_in, const int* in_sizes, int n_in,
                              void* d_out, int out_size, void* d_ws, size_t ws_size,
                              hipStream_t stream) {
    const float* x = (const float*)d_in[0];
    const float* Q = (const float*)d_in[1];
    const float* b = (const float*)d_in[2];
    const float* c = (const float*)d_in[3];
    float* out = (float*)d_out;

    const int K = in_sizes[1];            // 512
    const int N = out_size;               // 131072 rows
    const int nTiles = (N + 15) / 16;     // 16 rows per wave
    const int blocks = (nTiles + 7) / 8;  // 8 waves (256 threads) per block
    const size_t lds = 2 * (size_t)K * sizeof(float);  // Q + b staged in LDS

    DiagonalQuadratic_kernel<<<blocks, 256, lds, stream>>>(x, Q, b, c, out, nTiles, K);
}